// EdgeDecoder_9878424781118
// MI455X (gfx1250) — compile-verified
//
#include <hip/hip_runtime.h>
#include <hip/hip_bf16.h>
#include <math.h>

typedef __attribute__((ext_vector_type(2))) float v2f;
typedef __attribute__((ext_vector_type(8))) float v8f;

#define H0_STRIDE 260   // padded row: 260 % 64 == 4 -> conflict-free LDS columns, 16B-aligned rows
#define WAVES     4
#define PF        8     // B-fragment software-prefetch depth

// ---------------------------------------------------------------------------
// One-shot weight shuffle into exact WMMA B-fragment order:
//   wsB[( nt*64 + kk )*64 + L*2 + j] = W[kk*4 + 2*(L>>4) + j][nt*16 + (L&15)]
// so each B fragment is ONE fully-coalesced 256B chunk (lane L -> chunk+8L).
// wsB0: 16 ntiles * 64 kk * 64 floats ; wsB1: 8 ntiles * 64 kk * 64 floats
// ---------------------------------------------------------------------------
__global__ void shuffle_weights_kernel(const float* __restrict__ W0,   // [256][256] K-major
                                       const float* __restrict__ W1,   // [256][128] K-major
                                       float* __restrict__ wsB0,
                                       float* __restrict__ wsB1) {
    int i = blockIdx.x * 256 + threadIdx.x;          // 0 .. 65535
    {
        int j  = i & 1;
        int L  = (i >> 1) & 31;
        int kk = (i >> 6) & 63;
        int nt = i >> 12;                            // 0..15
        int k  = kk * 4 + 2 * (L >> 4) + j;
        int n  = nt * 16 + (L & 15);
        wsB0[i] = W0[(size_t)k * 256 + n];
    }
    if (i < 8 * 64 * 64) {
        int j  = i & 1;
        int L  = (i >> 1) & 31;
        int kk = (i >> 6) & 63;
        int nt = i >> 12;                            // 0..7
        int k  = kk * 4 + 2 * (L >> 4) + j;
        int n  = nt * 16 + (L & 15);
        wsB1[i] = W1[(size_t)k * 128 + n];
    }
}

// ---------------------------------------------------------------------------
// Main kernel: one wave owns a 16-edge M-tile. X and H0 live in registers as
// WMMA A-fragments (64 x v2f = 128 VGPRs); LDS only for the D->A shuffle.
// TW = weights pre-shuffled to fragment order in workspace.
// ---------------------------------------------------------------------------
template <bool TW>
__global__ __launch_bounds__(WAVES * 32)
void edge_decoder_wmma_kernel(const float* __restrict__ xc,   // [200000,128]
                              const float* __restrict__ xv,   // [50000,128]
                              const int*   __restrict__ row,  // [E]
                              const int*   __restrict__ col,  // [E]
                              const float* __restrict__ WB0,  // TW ? frag-order : [256][256] KN
                              const float* __restrict__ b0,   // [256]
                              const float* __restrict__ WB1,  // TW ? frag-order : [256][128] KN
                              const float* __restrict__ b1,   // [128]
                              const float* __restrict__ Wout, // [128]
                              const float* __restrict__ bout, // [1]
                              float* __restrict__ out,        // [E,2]
                              int E)
{
    __shared__ __align__(16) float smem[WAVES * 16 * H0_STRIDE];

    const int tid  = threadIdx.x;
    const int wave = tid >> 5;
    const int lane = tid & 31;
    float* H0s = smem + wave * 16 * H0_STRIDE;

    const int tileBase = (blockIdx.x * WAVES + wave) * 16;
    if (tileBase >= E) return;                       // wave-uniform exit

    const int mrow = lane & 15;                      // M (A) / N (B,C,D)
    const int h    = lane >> 4;                      // K-pair select: lanes>=16 hold K {2,3}

    // ---- gather layer-0 A fragments straight into registers (concat customer|variant) ----
    int e = tileBase + mrow; if (e > E - 1) e = E - 1;   // clamp, keep EXEC all-ones
    const float* crow = xc + (size_t)row[e] * 128 + 2 * h;
    const float* vrow = xv + (size_t)col[e] * 128 + 2 * h;

    v2f a[64];
    #pragma unroll
    for (int kk = 0; kk < 32; ++kk) a[kk]      = *(const v2f*)(crow + kk * 4);
    #pragma unroll
    for (int kk = 0; kk < 32; ++kk) a[kk + 32] = *(const v2f*)(vrow + kk * 4);

    // ---- layer0: [16x256] @ W0 + b0, leaky-relu -> H0s (A regs reused over 16 N-tiles) ----
    for (int nt = 0; nt < 16; ++nt) {
        int n = nt * 16 + mrow;
        float bias = b0[n];
        v8f c;
        #pragma unroll
        for (int j = 0; j < 8; ++j) c[j] = bias;
        if constexpr (TW) {
            const float* wp = WB0 + (size_t)nt * 64 * 64 + lane * 2;
            v2f bq[PF];
            #pragma unroll
            for (int i = 0; i < PF; ++i) bq[i] = *(const v2f*)(wp + i * 64);
            #pragma unroll
            for (int kk = 0; kk < 64; ++kk) {
                v2f bcur = bq[kk & (PF - 1)];
                if (kk + PF < 64)
                    bq[kk & (PF - 1)] = *(const v2f*)(wp + (kk + PF) * 64);
                c = __builtin_amdgcn_wmma_f32_16x16x4_f32(false, a[kk], false, bcur,
                                                          (short)0, c, false, false);
            }
        } else {
            #pragma unroll
            for (int kk = 0; kk < 64; ++kk) {
                const float* wp = WB0 + (size_t)(kk * 4 + 2 * h) * 256 + n;
                v2f b; b[0] = wp[0]; b[1] = wp[256];
                c = __builtin_amdgcn_wmma_f32_16x16x4_f32(false, a[kk], false, b,
                                                          (short)0, c, false, false);
            }
        }
        #pragma unroll
        for (int j = 0; j < 8; ++j) { float x = c[j]; c[j] = x > 0.f ? x : 0.01f * x; }
        #pragma unroll
        for (int j = 0; j < 8; ++j)                  // D: VGPR j -> row M = j + 8*h
            H0s[(j + 8 * h) * H0_STRIDE + nt * 16 + mrow] = c[j];
    }
    asm volatile("s_wait_dscnt 0" ::: "memory");     // wave-synchronous LDS hand-off

    // ---- reload H0 as A fragments into the same registers (once, reused over 8 N-tiles) ----
    #pragma unroll
    for (int kk = 0; kk < 64; ++kk)
        a[kk] = *(const v2f*)(H0s + mrow * H0_STRIDE + kk * 4 + 2 * h);

    // ---- layer1: [16x256] @ W1 + b1, leaky-relu -> H1 (overwrites H0s, cols 0..127) ----
    for (int nt = 0; nt < 8; ++nt) {
        int n = nt * 16 + mrow;
        float bias = b1[n];
        v8f c;
        #pragma unroll
        for (int j = 0; j < 8; ++j) c[j] = bias;
        if constexpr (TW) {
            const float* wp = WB1 + (size_t)nt * 64 * 64 + lane * 2;
            v2f bq[PF];
            #pragma unroll
            for (int i = 0; i < PF; ++i) bq[i] = *(const v2f*)(wp + i * 64);
            #pragma unroll
            for (int kk = 0; kk < 64; ++kk) {
                v2f bcur = bq[kk & (PF - 1)];
                if (kk + PF < 64)
                    bq[kk & (PF - 1)] = *(const v2f*)(wp + (kk + PF) * 64);
                c = __builtin_amdgcn_wmma_f32_16x16x4_f32(false, a[kk], false, bcur,
                                                          (short)0, c, false, false);
            }
        } else {
            #pragma unroll
            for (int kk = 0; kk < 64; ++kk) {
                const float* wp = WB1 + (size_t)(kk * 4 + 2 * h) * 128 + n;
                v2f b; b[0] = wp[0]; b[1] = wp[128];
                c = __builtin_amdgcn_wmma_f32_16x16x4_f32(false, a[kk], false, b,
                                                          (short)0, c, false, false);
            }
        }
        #pragma unroll
        for (int j = 0; j < 8; ++j) { float x = c[j]; c[j] = x > 0.f ? x : 0.01f * x; }
        #pragma unroll
        for (int j = 0; j < 8; ++j)
            H0s[(j + 8 * h) * H0_STRIDE + nt * 16 + mrow] = c[j];
    }
    asm volatile("s_wait_dscnt 0" ::: "memory");

    // ---- head: sigmoid(H1 . Wout + bout) -> [p, 1-p] (after all WMMAs; safe to predicate) ----
    if (lane < 16) {
        int eo = tileBase + lane;
        if (eo < E) {
            float acc = bout[0];
            const float* hrow = H0s + lane * H0_STRIDE;
            #pragma unroll 8
            for (int i = 0; i < 128; i += 4) {
                float4 hv = *(const float4*)(hrow + i);
                float4 wv = *(const float4*)(Wout + i);
                acc += hv.x * wv.x + hv.y * wv.y + hv.z * wv.z + hv.w * wv.w;
            }
            float p = 1.0f / (1.0f + __expf(-acc));
            out[2 * eo]     = p;
            out[2 * eo + 1] = 1.0f - p;
        }
    }
}

extern "C" void kernel_launch(void* const* d_in, const int* in_sizes, int n_in,
                              void* d_out, int out_size, void* d_ws, size_t ws_size,
                              hipStream_t stream) {
    const float* xc   = (const float*)d_in[0];
    const float* xv   = (const float*)d_in[1];
    const int*   row  = (const int*)  d_in[2];
    const int*   col  = (const int*)  d_in[3];
    const float* W0   = (const float*)d_in[4];
    const float* b0   = (const float*)d_in[5];
    const float* W1   = (const float*)d_in[6];
    const float* b1   = (const float*)d_in[7];
    const float* Wout = (const float*)d_in[8];
    const float* bout = (const float*)d_in[9];
    float* out = (float*)d_out;

    int E = in_sizes[2];
    int edgesPerBlock = WAVES * 16;                  // 64
    int grid = (E + edgesPerBlock - 1) / edgesPerBlock;

    size_t need = (size_t)(16 * 64 * 64 + 8 * 64 * 64) * sizeof(float);  // 384 KB
    if (ws_size >= need) {
        float* wsB0 = (float*)d_ws;
        float* wsB1 = wsB0 + 16 * 64 * 64;
        shuffle_weights_kernel<<<256, 256, 0, stream>>>(W0, W1, wsB0, wsB1);
        edge_decoder_wmma_kernel<true><<<grid, WAVES * 32, 0, stream>>>(
            xc, xv, row, col, wsB0, b0, wsB1, b1, Wout, bout, out, E);
    } else {
        edge_decoder_wmma_kernel<false><<<grid, WAVES * 32, 0, stream>>>(
            xc, xv, row, col, W0, b0, W1, b1, Wout, bout, out, E);
    }
}